// GIN_21045339750447
// MI455X (gfx1250) — compile-verified
//
#include <hip/hip_runtime.h>

#define N_NODES 50000
#define N_EDGES 800000
#define DH 128
#define D_OUT 64
#define N_CONV 4

typedef __attribute__((ext_vector_type(2))) float v2f;
typedef __attribute__((ext_vector_type(8))) float v8f;

// -------- edge scatter: out[dst] += h[src], per-edge float4 chunks, HW f32 atomics
__global__ __launch_bounds__(256) void gin_scatter(const float* __restrict__ h,
                                                   const int* __restrict__ src,
                                                   const int* __restrict__ dst,
                                                   float* __restrict__ out) {
  const int tid = blockIdx.x * 256 + threadIdx.x;
  const int e = tid >> 5;               // 32 chunks of 4 floats = 128 feats
  if (e >= N_EDGES) return;
  const int c4 = (tid & 31) * 4;
  const float4 v = *(const float4*)(h + (size_t)src[e] * DH + c4);
  float* p = out + (size_t)dst[e] * DH + c4;
  __hip_atomic_fetch_add(p + 0, v.x, __ATOMIC_RELAXED, __HIP_MEMORY_SCOPE_AGENT);
  __hip_atomic_fetch_add(p + 1, v.y, __ATOMIC_RELAXED, __HIP_MEMORY_SCOPE_AGENT);
  __hip_atomic_fetch_add(p + 2, v.z, __ATOMIC_RELAXED, __HIP_MEMORY_SCOPE_AGENT);
  __hip_atomic_fetch_add(p + 3, v.w, __ATOMIC_RELAXED, __HIP_MEMORY_SCOPE_AGENT);
}

// -------- C[M x 128] = A[M x 128] @ W[128 x 128] via V_WMMA_F32_16X16X4_F32
// one wave per 16x16 output tile; 8 waves/block cover 128 rows; grid.x = 8 col tiles
__global__ __launch_bounds__(256) void gemm128_wmma(const float* __restrict__ A,
                                                    const float* __restrict__ W,
                                                    float* __restrict__ C,
                                                    int nrows) {
  const int lane = threadIdx.x & 31;
  const int wave = threadIdx.x >> 5;
  const int half = lane >> 4;           // 0 => lanes 0-15, 1 => lanes 16-31
  const int l16  = lane & 15;
  const int n0   = blockIdx.x * 16;
  const int m0   = (blockIdx.y * 8 + wave) * 16;

  int arow = m0 + l16;
  if (arow >= nrows) arow = nrows - 1;  // clamp; tile tail stores are guarded
  const float* __restrict__ Arow = A + (size_t)arow * DH;

  v8f acc = {};
#pragma unroll
  for (int kb = 0; kb < 32; ++kb) {
    const int kk = kb * 4 + half * 2;   // A lanes 0-15: K=0,1 ; lanes 16-31: K=2,3
    v2f a = *(const v2f*)(Arow + kk);   // 8B-aligned (kk even)
    v2f b;
    b.x = W[(size_t)kk * DH + n0 + l16];        // B VGPR0: K=kk row
    b.y = W[(size_t)(kk + 1) * DH + n0 + l16];  // B VGPR1: K=kk+1 row
    acc = __builtin_amdgcn_wmma_f32_16x16x4_f32(false, a, false, b, (short)0, acc,
                                                false, false);
  }
#pragma unroll
  for (int r = 0; r < 8; ++r) {         // C/D: VGPR r -> M = r + 8*half
    const int row = m0 + r + half * 8;
    if (row < nrows) C[(size_t)row * DH + n0 + l16] = acc[r];
  }
}

// -------- per-channel batch stats -> mean, rstd (population variance, BN eps)
__global__ __launch_bounds__(256) void bn_stats(const float* __restrict__ x,
                                                float* __restrict__ mean,
                                                float* __restrict__ rstd) {
  __shared__ float s1[256], s2[256];
  const int c = blockIdx.x;
  float sum = 0.f, sq = 0.f;
  for (int r = threadIdx.x; r < N_NODES; r += 256) {
    const float v = x[(size_t)r * DH + c];
    sum += v; sq += v * v;
  }
  s1[threadIdx.x] = sum; s2[threadIdx.x] = sq;
  __syncthreads();
  for (int s = 128; s > 0; s >>= 1) {
    if (threadIdx.x < s) { s1[threadIdx.x] += s1[threadIdx.x + s];
                           s2[threadIdx.x] += s2[threadIdx.x + s]; }
    __syncthreads();
  }
  if (threadIdx.x == 0) {
    const float m = s1[0] * (1.f / N_NODES);
    const float var = s2[0] * (1.f / N_NODES) - m * m;
    mean[c] = m;
    rstd[c] = rsqrtf(var + 1e-5f);
  }
}

// -------- in-place BN + ReLU
__global__ __launch_bounds__(256) void bn_relu(float* __restrict__ x,
                                               const float* __restrict__ mean,
                                               const float* __restrict__ rstd,
                                               const float* __restrict__ g,
                                               const float* __restrict__ b) {
  const size_t i = (size_t)blockIdx.x * 256 + threadIdx.x;
  if (i >= (size_t)N_NODES * DH) return;
  const int c = (int)(i & (DH - 1));
  const float v = (x[i] - mean[c]) * rstd[c] * g[c] + b[c];
  x[i] = v > 0.f ? v : 0.f;
}

// -------- column (per-channel) sum over nodes -> pooled[128]
__global__ __launch_bounds__(256) void col_sum(const float* __restrict__ x,
                                               float* __restrict__ pooled) {
  __shared__ float s1[256];
  const int c = blockIdx.x;
  float sum = 0.f;
  for (int r = threadIdx.x; r < N_NODES; r += 256)
    sum += x[(size_t)r * DH + c];
  s1[threadIdx.x] = sum;
  __syncthreads();
  for (int s = 128; s > 0; s >>= 1) {
    if (threadIdx.x < s) s1[threadIdx.x] += s1[threadIdx.x + s];
    __syncthreads();
  }
  if (threadIdx.x == 0) pooled[c] = s1[0];
}

// -------- 5 prediction heads: score = sum_i pooled[i] @ predW[i] + predb[i]
__global__ __launch_bounds__(64) void heads(const float* __restrict__ pooled,  // [5][128]
                                            const float* __restrict__ predW,   // [5][128][64]
                                            const float* __restrict__ predb,   // [5][64]
                                            float* __restrict__ out) {
  const int o = threadIdx.x;
  if (o >= D_OUT) return;
  float acc = 0.f;
  for (int i = 0; i <= N_CONV; ++i) {
    float s = predb[i * D_OUT + o];
    const float* Wp = predW + (size_t)i * DH * D_OUT;
    const float* pp = pooled + i * DH;
    for (int c = 0; c < DH; ++c)
      s += pp[c] * Wp[c * D_OUT + o];
    acc += s;
  }
  out[o] = acc;
}

extern "C" void kernel_launch(void* const* d_in, const int* in_sizes, int n_in,
                              void* d_out, int out_size, void* d_ws, size_t ws_size,
                              hipStream_t stream) {
  (void)in_sizes; (void)n_in; (void)out_size; (void)ws_size;
  const float* h_in  = (const float*)d_in[0];
  const int*   src   = (const int*)d_in[1];
  const int*   dst   = (const int*)d_in[2];
  const float* W1    = (const float*)d_in[3];   // [4][128][128]
  const float* W2    = (const float*)d_in[4];   // [4][128][128]
  const float* bn1g  = (const float*)d_in[5];
  const float* bn1b  = (const float*)d_in[6];
  const float* bn2g  = (const float*)d_in[7];
  const float* bn2b  = (const float*)d_in[8];
  const float* predW = (const float*)d_in[9];   // [5][128][64]
  const float* predb = (const float*)d_in[10];  // [5][64]

  const size_t nodeElems = (size_t)N_NODES * DH;
  float* bufZ   = (float*)d_ws;
  float* bufT   = bufZ + nodeElems;
  float* bufH   = bufT + nodeElems;
  float* pooled = bufH + nodeElems;             // [5][128]
  float* mean   = pooled + (N_CONV + 1) * DH;   // [128]
  float* rstd   = mean + DH;                    // [128]

  const dim3 gemmGrid(DH / 16, (N_NODES + 127) / 128);
  const int scatBlocks = (N_EDGES * 32 + 255) / 256;
  const int eltBlocks  = (int)((nodeElems + 255) / 256);

  // hidden_rep[0]: pooled input features
  col_sum<<<128, 256, 0, stream>>>(h_in, pooled);

  const float* hc = h_in;
  for (int i = 0; i < N_CONV; ++i) {
    // z = h + segment_sum(h[src], dst)
    hipMemcpyAsync(bufZ, hc, nodeElems * sizeof(float),
                   hipMemcpyDeviceToDevice, stream);
    gin_scatter<<<scatBlocks, 256, 0, stream>>>(hc, src, dst, bufZ);
    // z @ W1 -> BN -> ReLU
    gemm128_wmma<<<gemmGrid, 256, 0, stream>>>(bufZ, W1 + (size_t)i * DH * DH,
                                               bufT, N_NODES);
    bn_stats<<<128, 256, 0, stream>>>(bufT, mean, rstd);
    bn_relu<<<eltBlocks, 256, 0, stream>>>(bufT, mean, rstd,
                                           bn1g + i * DH, bn1b + i * DH);
    // @ W2 -> BN -> ReLU = next h
    gemm128_wmma<<<gemmGrid, 256, 0, stream>>>(bufT, W2 + (size_t)i * DH * DH,
                                               bufH, N_NODES);
    bn_stats<<<128, 256, 0, stream>>>(bufH, mean, rstd);
    bn_relu<<<eltBlocks, 256, 0, stream>>>(bufH, mean, rstd,
                                           bn2g + i * DH, bn2b + i * DH);
    col_sum<<<128, 256, 0, stream>>>(bufH, pooled + (i + 1) * DH);
    hc = bufH;
  }

  heads<<<1, 64, 0, stream>>>(pooled, predW, predb, (float*)d_out);
}